// SGC_33801392619927
// MI455X (gfx1250) — compile-verified
//
#include <hip/hip_runtime.h>
#include <cstdint>

typedef __attribute__((ext_vector_type(2))) float v2f;
typedef __attribute__((ext_vector_type(8))) float v8f;

#define F_IN  128
#define F_HID 64
#define F_OUT 40

// ---------------------------------------------------------------------------
// 1) GCN normalization: deg (with self loop) -> dis = rsqrt(deg)
// ---------------------------------------------------------------------------
__global__ void sgc_init_deg(float* deg, int n) {
    int i = blockIdx.x * blockDim.x + threadIdx.x;
    if (i < n) deg[i] = 1.0f;  // self loop
}

__global__ void sgc_count_deg(float* deg, const int64_t* dst, long long e) {
    long long i = (long long)blockIdx.x * blockDim.x + threadIdx.x;
    if (i < e) atomicAdd(&deg[(int)dst[i]], 1.0f);
}

__global__ void sgc_finish_dis(float* dis, int n) {
    int i = blockIdx.x * blockDim.x + threadIdx.x;
    if (i < n) {
        float d = dis[i];
        dis[i] = (d > 0.0f) ? rsqrtf(d) : 0.0f;
    }
}

// ---------------------------------------------------------------------------
// 2) Propagation hop: out = A_hat_norm @ in   (feature dim = 1<<fshift).
//    Self-loop term initializes out; then edge-parallel scatter with one
//    thread per (edge, float4): 128-bit gathers, 4x less index redundancy.
//    Destination rows (<= 51MB) are L2-resident -> atomics resolve in L2.
// ---------------------------------------------------------------------------
__global__ void sgc_self_init4(float* __restrict__ out,
                               const float* __restrict__ in,
                               const float* __restrict__ dis,
                               int n, int fshift) {
    long long idx = (long long)blockIdx.x * blockDim.x + threadIdx.x;  // float4 idx
    long long total4 = (long long)n << (fshift - 2);
    if (idx >= total4) return;
    long long i = idx >> (fshift - 2);
    float w = dis[i];
    w = w * w;
    float4 v = ((const float4*)in)[idx];
    float4 r;
    r.x = w * v.x; r.y = w * v.y; r.z = w * v.z; r.w = w * v.w;
    ((float4*)out)[idx] = r;
}

__global__ void sgc_scatter4(float* __restrict__ out,
                             const float* __restrict__ in,
                             const float* __restrict__ dis,
                             const int64_t* __restrict__ src,
                             const int64_t* __restrict__ dst,
                             long long e, int fshift) {
    long long tid = (long long)blockIdx.x * blockDim.x + threadIdx.x;
    int tpe_shift = fshift - 2;                 // threads per edge = Fd/4
    long long eid = tid >> tpe_shift;
    if (eid >= e) return;
    int f4 = (int)(tid & ((1LL << tpe_shift) - 1));
    int s = (int)src[eid];
    int d = (int)dst[eid];
    // Prefetch next edge's source row (global_prefetch_b8 on gfx1250).
    if (eid + 1 < e) {
        int sn = (int)src[eid + 1];
        __builtin_prefetch(&in[((long long)sn << fshift) + (f4 << 2)], 0, 1);
    }
    float w = dis[s] * dis[d];
    float4 v = ((const float4*)(in + ((long long)s << fshift)))[f4];
    float* o = out + ((long long)d << fshift) + (f4 << 2);
    atomicAdd(o + 0, w * v.x);
    atomicAdd(o + 1, w * v.y);
    atomicAdd(o + 2, w * v.z);
    atomicAdd(o + 3, w * v.w);
}

// ---------------------------------------------------------------------------
// 3) GEMM1 fused bias+ReLU via V_WMMA_F32_16X16X4_F32. Block = 128 thr =
//    4 waves; wave w owns the 16x16 tile at col 16*w. The shared 16x128
//    A-tile is staged once in LDS (stride 132 floats: 8B-aligned, and the
//    16-lane stride-132 reads map to distinct banks -> conflict-free).
// ---------------------------------------------------------------------------
__global__ void sgc_gemm1_relu(const float* __restrict__ A,   // N x 128
                               const float* __restrict__ W,   // 128 x 64
                               const float* __restrict__ bias,
                               float* __restrict__ out,       // N x 64
                               int n) {
    __shared__ float As[16 * 132];
    int tid = threadIdx.x;
    int rowBase = blockIdx.x * 16;
    {
        int r = tid >> 3;      // 0..15
        int sgm = tid & 7;     // 8 segments x 16 floats
        const float4* g = (const float4*)(A + (long long)(rowBase + r) * F_IN + sgm * 16);
        float4* l = (float4*)(&As[r * 132 + sgm * 16]);
#pragma unroll
        for (int i = 0; i < 4; i++) l[i] = g[i];
    }
    __syncthreads();

    int wave = tid >> 5;
    int lane = tid & 31;
    int colBase = wave * 16;
    int lrow  = lane & 15;
    int khalf = (lane >> 4) * 2;   // lanes 0-15: (K0,K1); lanes 16-31: (K2,K3)
    int col   = colBase + lrow;
    const float* arow = &As[lrow * 132];

    v8f acc = {};
    for (int k = 0; k < F_IN; k += 4) {
        v2f a, b;
        a.x = arow[k + khalf];
        a.y = arow[k + khalf + 1];
        b.x = W[(k + khalf) * F_HID + col];
        b.y = W[(k + khalf + 1) * F_HID + col];
        acc = __builtin_amdgcn_wmma_f32_16x16x4_f32(
            false, a, false, b, (short)0, acc, false, false);
    }

    float bv = bias[col];
    int rowOff = (lane >> 4) * 8;  // C/D layout: lanes 16-31 hold M = i+8
#pragma unroll
    for (int i = 0; i < 8; i++) {
        int row = rowBase + rowOff + i;
        if (row < n)
            out[(long long)row * F_HID + col] = fmaxf(acc[i] + bv, 0.0f);
    }
}

// ---------------------------------------------------------------------------
// 4) GEMM2 (64 -> 40) + bias via f32 WMMA. Block = 96 thr = 3 waves over 3
//    column tiles (48 wide, cols >= 40 masked on B-loads/stores only, so
//    EXEC stays all-1s through the WMMA). A-tile staged in LDS (stride 68).
// ---------------------------------------------------------------------------
__global__ void sgc_gemm2(const float* __restrict__ A,   // N x 64
                          const float* __restrict__ W,   // 64 x 40
                          const float* __restrict__ bias,
                          float* __restrict__ out,       // N x 40 (logits)
                          int n) {
    __shared__ float As[16 * 68];
    int tid = threadIdx.x;
    int rowBase = blockIdx.x * 16;
    for (int idx = tid; idx < 16 * F_HID; idx += 96) {
        int r = idx >> 6, c = idx & 63;
        As[r * 68 + c] = A[(long long)(rowBase + r) * F_HID + c];
    }
    __syncthreads();

    int wave = tid >> 5;   // 0..2
    int lane = tid & 31;
    int colBase = wave * 16;
    int lrow  = lane & 15;
    int khalf = (lane >> 4) * 2;
    int col   = colBase + lrow;
    bool cok  = (col < F_OUT);
    const float* arow = &As[lrow * 68];

    v8f acc = {};
    for (int k = 0; k < F_HID; k += 4) {
        v2f a, b;
        a.x = arow[k + khalf];
        a.y = arow[k + khalf + 1];
        b.x = cok ? W[(k + khalf) * F_OUT + col] : 0.0f;
        b.y = cok ? W[(k + khalf + 1) * F_OUT + col] : 0.0f;
        acc = __builtin_amdgcn_wmma_f32_16x16x4_f32(
            false, a, false, b, (short)0, acc, false, false);
    }

    if (cok) {
        float bv = bias[col];
        int rowOff = (lane >> 4) * 8;
#pragma unroll
        for (int i = 0; i < 8; i++) {
            int row = rowBase + rowOff + i;
            if (row < n) out[(long long)row * F_OUT + col] = acc[i] + bv;
        }
    }
}

// ---------------------------------------------------------------------------
// 5) Row-wise log_softmax over 40 logits. Output is write-once -> NT stores.
// ---------------------------------------------------------------------------
__global__ void sgc_log_softmax(const float* __restrict__ logits,
                                float* __restrict__ out, int n) {
    int i = blockIdx.x * blockDim.x + threadIdx.x;
    if (i >= n) return;
    const float* r = logits + (long long)i * F_OUT;
    float m = r[0];
#pragma unroll
    for (int c = 1; c < F_OUT; c++) m = fmaxf(m, r[c]);
    float s = 0.0f;
#pragma unroll
    for (int c = 0; c < F_OUT; c++) s += __expf(r[c] - m);
    float lse = m + __logf(s);
    float* o = out + (long long)i * F_OUT;
#pragma unroll
    for (int c = 0; c < F_OUT; c++)
        __builtin_nontemporal_store(r[c] - lse, &o[c]);
}

// ---------------------------------------------------------------------------
// Orchestration
// ---------------------------------------------------------------------------
extern "C" void kernel_launch(void* const* d_in, const int* in_sizes, int n_in,
                              void* d_out, int out_size, void* d_ws, size_t ws_size,
                              hipStream_t stream) {
    const float*   x   = (const float*)d_in[0];
    const int64_t* ei  = (const int64_t*)d_in[1];
    const float*   W1  = (const float*)d_in[2];
    const float*   b1  = (const float*)d_in[3];
    const float*   W2  = (const float*)d_in[4];
    const float*   b2  = (const float*)d_in[5];
    float*         out = (float*)d_out;

    const int       N = in_sizes[0] / F_IN;      // 100000
    const long long E = in_sizes[1] / 2;         // 1600000
    const int64_t* src = ei;
    const int64_t* dst = ei + E;

    // Workspace layout (floats): dis[N] | bufA[N*128] | bufB[N*128]
    float* dis  = (float*)d_ws;
    float* bufA = dis + N;
    float* bufB = bufA + (size_t)N * F_IN;

    const int TB = 256;
    const unsigned gN    = (unsigned)((N + TB - 1) / TB);
    const unsigned gE    = (unsigned)((E + TB - 1) / TB);
    const unsigned gNF4  = (unsigned)((((long long)N << 5) + TB - 1) / TB);  // N*128/4
    const unsigned gEF4  = (unsigned)((((long long)E << 5) + TB - 1) / TB);  // E*128/4
    const unsigned gNH4  = (unsigned)((((long long)N << 4) + TB - 1) / TB);  // N*64/4
    const unsigned gEH4  = (unsigned)((((long long)E << 4) + TB - 1) / TB);  // E*64/4
    const unsigned gTiles = (unsigned)((N + 15) / 16);

    // Normalization
    sgc_init_deg  <<<gN, TB, 0, stream>>>(dis, N);
    sgc_count_deg <<<gE, TB, 0, stream>>>(dis, dst, E);
    sgc_finish_dis<<<gN, TB, 0, stream>>>(dis, N);

    // Layer 1: two propagation hops at F=128 (fshift=7)
    sgc_self_init4<<<gNF4, TB, 0, stream>>>(bufA, x, dis, N, 7);
    sgc_scatter4  <<<gEF4, TB, 0, stream>>>(bufA, x, dis, src, dst, E, 7);
    sgc_self_init4<<<gNF4, TB, 0, stream>>>(bufB, bufA, dis, N, 7);
    sgc_scatter4  <<<gEF4, TB, 0, stream>>>(bufB, bufA, dis, src, dst, E, 7);

    // Linear 1 + ReLU (WMMA f32): bufB (N x 128) -> bufA (N x 64)
    sgc_gemm1_relu<<<gTiles, 128, 0, stream>>>(bufB, W1, b1, bufA, N);

    // Layer 2: two propagation hops at F=64 (fshift=6)
    sgc_self_init4<<<gNH4, TB, 0, stream>>>(bufB, bufA, dis, N, 6);
    sgc_scatter4  <<<gEH4, TB, 0, stream>>>(bufB, bufA, dis, src, dst, E, 6);
    sgc_self_init4<<<gNH4, TB, 0, stream>>>(bufA, bufB, dis, N, 6);
    sgc_scatter4  <<<gEH4, TB, 0, stream>>>(bufA, bufB, dis, src, dst, E, 6);

    // Linear 2 (WMMA f32): bufA (N x 64) -> logits bufB (N x 40)
    sgc_gemm2<<<gTiles, 96, 0, stream>>>(bufA, W2, b2, bufB, N);

    // log_softmax -> d_out
    sgc_log_softmax<<<gN, TB, 0, stream>>>(bufB, out, N);
}